// Jac_46042049413373
// MI455X (gfx1250) — compile-verified
//
#include <hip/hip_runtime.h>
#include <stdint.h>

// ---------------------------------------------------------------------------
// CDNA5 async global->LDS support (gfx1250). Guarded so the file compiles on
// any toolchain; when available we emit global_load_async_to_lds_b32 +
// s_wait_asynccnt (tracked by ASYNCcnt per CDNA5 ISA).
// Probe round 1 showed the builtin signature takes typed pointers:
//   (addrspace(1) int*, addrspace(3) int*, imm offset, imm cpol)
// ---------------------------------------------------------------------------
#if defined(__has_builtin)
#if __has_builtin(__builtin_amdgcn_global_load_async_to_lds_b32)
#define HAVE_ASYNC_LDS 1
#endif
#endif
#ifndef HAVE_ASYNC_LDS
#define HAVE_ASYNC_LDS 0
#endif

#if HAVE_ASYNC_LDS
typedef __attribute__((address_space(1))) int g_int;
typedef __attribute__((address_space(3))) int l_int;
#endif

#define LIST_CAP 1024   // per-list LDS capacity (ints); lists average ~32

__device__ __forceinline__ int lower_bound_i32(const int* __restrict__ a, int n, int key) {
    int lo = 0, hi = n;
    while (lo < hi) {
        int mid = (lo + hi) >> 1;
        if (a[mid] < key) lo = mid + 1; else hi = mid;
    }
    return lo;
}

// slot[n] = 2*nb for all nodes; counts[i] = 0 for i < 2*nb
__global__ void k_init(int* __restrict__ slot, int* __restrict__ counts, int N, int nb) {
    int i = blockIdx.x * blockDim.x + threadIdx.x;
    if (i < N) slot[i] = 2 * nb;
    if (i < 2 * nb) counts[i] = 0;
}

// slot[searchsorted(batch, g)] = g        (first node of graph g)
__global__ void k_slot_src(const int* __restrict__ batch, int N, int nb, int* __restrict__ slot) {
    int g = blockIdx.x * blockDim.x + threadIdx.x;
    if (g >= nb) return;
    int idx = lower_bound_i32(batch, N, g);
    if (idx < N) slot[idx] = g;           // OOB scatter dropped, as in JAX
}

// slot[searchsorted(batch, g) + 1] = g + nb   (second node; applied after src)
__global__ void k_slot_dst(const int* __restrict__ batch, int N, int nb, int* __restrict__ slot) {
    int g = blockIdx.x * blockDim.x + threadIdx.x;
    if (g >= nb) return;
    int idx = lower_bound_i32(batch, N, g) + 1;
    if (idx < N) slot[idx] = g + nb;
}

// counts[slot[u]]++ for relevant edges
__global__ void k_count(const int* __restrict__ ei0, int E, const int* __restrict__ slot,
                        int nb, int* __restrict__ counts) {
    int e = blockIdx.x * blockDim.x + threadIdx.x;
    if (e >= E) return;
    // stream prefetch of the edge array ahead (global_prefetch_b8, speculative)
    if (e + 4096 < E) __builtin_prefetch(ei0 + e + 4096, 0, 0);
    int s = slot[ei0[e]];
    if (s < 2 * nb) atomicAdd(&counts[s], 1);
}

// serial exclusive scan over 2*nb counters (2000 elements -> negligible)
__global__ void k_scan(const int* __restrict__ counts, int n2,
                       int* __restrict__ offsets, int* __restrict__ cursors) {
    if (blockIdx.x == 0 && threadIdx.x == 0) {
        int acc = 0;
        for (int i = 0; i < n2; ++i) {
            offsets[i] = acc;
            cursors[i] = acc;
            acc += counts[i];
        }
        offsets[n2] = acc;
    }
}

// bucket[cursor[slot[u]]++] = v for relevant edges
__global__ void k_scatter(const int* __restrict__ ei0, const int* __restrict__ ei1, int E,
                          const int* __restrict__ slot, int nb,
                          int* __restrict__ cursors, int* __restrict__ bucket) {
    int e = blockIdx.x * blockDim.x + threadIdx.x;
    if (e >= E) return;
    if (e + 4096 < E) __builtin_prefetch(ei1 + e + 4096, 0, 0);
    int s = slot[ei0[e]];
    if (s < 2 * nb) {
        int p = atomicAdd(&cursors[s], 1);
        bucket[p] = ei1[e];
    }
}

// One wave32 per graph: jac[g] = pairs(A,B) / distinct(A ∪ B)
__global__ __launch_bounds__(32) void k_jaccard(const int* __restrict__ bucket,
                                                const int* __restrict__ offsets,
                                                int nb, float* __restrict__ out) {
    __shared__ int shA[LIST_CAP];
    __shared__ int shB[LIST_CAP];

    int g = blockIdx.x;
    if (g >= nb) return;
    int lane = threadIdx.x;

    int a0 = offsets[g],      a1 = offsets[g + 1];
    int b0 = offsets[nb + g], b1 = offsets[nb + g + 1];
    int m = a1 - a0, n = b1 - b0;

    const int* A;
    const int* B;
    bool use_lds = (m <= LIST_CAP) && (n <= LIST_CAP);
    if (use_lds) {
        for (int k = lane; k < m; k += 32) {
#if HAVE_ASYNC_LDS
            __builtin_amdgcn_global_load_async_to_lds_b32(
                (g_int*)(bucket + a0 + k), (l_int*)&shA[k], 0, 0);
#else
            shA[k] = bucket[a0 + k];
#endif
        }
        for (int k = lane; k < n; k += 32) {
#if HAVE_ASYNC_LDS
            __builtin_amdgcn_global_load_async_to_lds_b32(
                (g_int*)(bucket + b0 + k), (l_int*)&shB[k], 0, 0);
#else
            shB[k] = bucket[b0 + k];
#endif
        }
#if HAVE_ASYNC_LDS
        asm volatile("s_wait_asynccnt 0" ::: "memory");
#endif
        __syncthreads();
        A = shA;
        B = shB;
    } else {                      // overflow fallback (not expected with this input)
        A = bucket + a0;
        B = bucket + b0;
    }

    // cn = #pairs (i,j) with A[i]==B[j]  == sum_v cntA(v)*cntB(v)
    int cn = 0;
    for (int i = lane; i < m; i += 32) {
        int av = A[i];
        int c = 0;
        for (int j = 0; j < n; ++j) c += (B[j] == av) ? 1 : 0;
        cn += c;
    }

    // union = #distinct values in concat(A,B): count first occurrences
    int L = m + n;
    int uni = 0;
    for (int i = lane; i < L; i += 32) {
        int v = (i < m) ? A[i] : B[i - m];
        bool first = true;
        for (int j = 0; j < i; ++j) {
            int w = (j < m) ? A[j] : B[j - m];
            if (w == v) { first = false; break; }
        }
        uni += first ? 1 : 0;
    }

    // wave32 reduction
    for (int off = 16; off > 0; off >>= 1) {
        cn  += __shfl_down(cn,  off, 32);
        uni += __shfl_down(uni, off, 32);
    }
    if (lane == 0) out[g] = (uni > 0) ? ((float)cn / (float)uni) : 0.0f;
}

extern "C" void kernel_launch(void* const* d_in, const int* in_sizes, int n_in,
                              void* d_out, int out_size, void* d_ws, size_t ws_size,
                              hipStream_t stream) {
    // inputs: z(float,N*16) [unused], edge_index(int,2*E), batch(int,N), num_graphs(int,1)
    const int* edge  = (const int*)d_in[1];
    const int* batch = (const int*)d_in[2];
    const int  N  = in_sizes[2];
    const int  E  = in_sizes[1] / 2;
    const int  nb = out_size;            // out has one float per graph
    float* out = (float*)d_out;

    // workspace layout (ints): slot[N] | counts[2nb] | offsets[2nb+1] | cursors[2nb] | bucket[E]
    int* W       = (int*)d_ws;
    int* slot    = W;
    int* counts  = slot + N;
    int* offsets = counts + 2 * nb;
    int* cursors = offsets + (2 * nb + 1);
    int* bucket  = cursors + 2 * nb;

    const int T = 256;
    int initN = (N > 2 * nb) ? N : 2 * nb;
    k_init   <<<(initN + T - 1) / T, T, 0, stream>>>(slot, counts, N, nb);
    k_slot_src<<<(nb + T - 1) / T, T, 0, stream>>>(batch, N, nb, slot);
    k_slot_dst<<<(nb + T - 1) / T, T, 0, stream>>>(batch, N, nb, slot);
    k_count  <<<(E + T - 1) / T, T, 0, stream>>>(edge, E, slot, nb, counts);
    k_scan   <<<1, 1, 0, stream>>>(counts, 2 * nb, offsets, cursors);
    k_scatter<<<(E + T - 1) / T, T, 0, stream>>>(edge, edge + E, E, slot, nb, cursors, bucket);
    k_jaccard<<<nb, 32, 0, stream>>>(bucket, offsets, nb, out);
}